// GSModel_14851996910207
// MI455X (gfx1250) — compile-verified
//
#include <hip/hip_runtime.h>
#include <math.h>

#define NPTS 4096
#define HIMG 96
#define WIMG 96
#define NPIX (HIMG * WIMG)   // 9216
#define RECF 12              // floats per packed gaussian record
#define WSTRIDE 48           // LDS row stride for weight tiles (bank-conflict free)

typedef __attribute__((ext_vector_type(2))) float v2f;
typedef __attribute__((ext_vector_type(8))) float v8f;
typedef __attribute__((ext_vector_type(4))) unsigned v4u;
typedef __attribute__((ext_vector_type(8))) unsigned v8u;

// low 32 bits of a generic pointer to __shared__ = LDS byte offset
__device__ __forceinline__ unsigned lds_off_u32(const void* p) {
    return (unsigned)(unsigned long long)p;
}

__device__ __forceinline__ void wait_tensorcnt0() {
#if __has_builtin(__builtin_amdgcn_s_wait_tensorcnt)
    __builtin_amdgcn_s_wait_tensorcnt(0);
#else
    asm volatile("s_wait_tensorcnt 0" ::: "memory");
#endif
}

__device__ __forceinline__ void wait_asynccnt0() {
#if __has_builtin(__builtin_amdgcn_s_wait_asynccnt)
    __builtin_amdgcn_s_wait_asynccnt(0);
#else
    asm volatile("s_wait_asynccnt 0" ::: "memory");
#endif
}

// TDM: DMA one 192-float (768 B) record chunk from global into LDS.
// D# per CDNA5 ISA:
//   group0: count=1 | lds_addr | global_addr[56:0] | type=2
//   group1: data_size=4B, tensor_dim0=49152, tensor_dim1=1,
//           tile_dim0=192, tile_dim1=1, tensor_dim0_stride=49152
__device__ __forceinline__ void tdm_load_chunk(unsigned ldsDst, const float* gsrc) {
    const unsigned long long ga = (unsigned long long)gsrc;
    v4u g0;
    g0.x = 1u;                                                   // count=1 (valid)
    g0.y = ldsDst;                                               // lds_addr
    g0.z = (unsigned)ga;                                         // global_addr[31:0]
    g0.w = (unsigned)((ga >> 32) & 0x1FFFFFFull) | 0x80000000u;  // [56:32] | type=2
    v8u g1;
    g1.s0 = 0x00020000u;      // workgroup_mask=0, data_size=2 (4 bytes), no flags
    g1.s1 = 0xC0000000u;      // tensor_dim0 = 49152 (bits 79:48, low half here)
    g1.s2 = 0x00010000u;      // tensor_dim0 hi = 0, tensor_dim1 = 1
    g1.s3 = (192u << 16);     // tile_dim0 = 192 (bits 127:112)
    g1.s4 = 1u;               // tile_dim1 = 1, tile_dim2 = 0
    g1.s5 = 49152u;           // tensor_dim0_stride low 32
    g1.s6 = 0u;
    g1.s7 = 0u;
    asm volatile("tensor_load_to_lds %0, %1" :: "s"(g0), "s"(g1) : "memory");
}

// ---------------------------------------------------------------------------
// Kernel 1: per-Gaussian preprocessing.
// Writes: rec[n*12] = {u0,u1,cinvA,cinvB,cinvC,alphaEff,r,g,b,pad,pad,pad},
//         depth[n], areas (directly into d_out, original order).
// ---------------------------------------------------------------------------
__global__ void __launch_bounds__(256) preprocess_kernel(
    const float* __restrict__ pws, const float* __restrict__ low_shs,
    const float* __restrict__ high_shs, const float* __restrict__ alphas_raw,
    const float* __restrict__ scales_raw, const float* __restrict__ rots_raw,
    const float* __restrict__ Rcw, const float* __restrict__ tcw,
    const float* __restrict__ cam, float* __restrict__ rec,
    float* __restrict__ depthOut, float* __restrict__ areasOut)
{
    int n = blockIdx.x * 256 + threadIdx.x;
    if (n >= NPTS) return;

    const float fx = cam[0], fy = cam[1], cx = cam[2], cy = cam[3];
    const float R00 = Rcw[0], R01 = Rcw[1], R02 = Rcw[2];
    const float R10 = Rcw[3], R11 = Rcw[4], R12 = Rcw[5];
    const float R20 = Rcw[6], R21 = Rcw[7], R22 = Rcw[8];
    const float t0 = tcw[0], t1 = tcw[1], t2 = tcw[2];

    const float pwx = pws[n * 3 + 0], pwy = pws[n * 3 + 1], pwz = pws[n * 3 + 2];

    // camera-space position
    const float pcx = R00 * pwx + R01 * pwy + R02 * pwz + t0;
    const float pcy = R10 * pwx + R11 * pwy + R12 * pwz + t1;
    const float pcz = R20 * pwx + R21 * pwy + R22 * pwz + t2;
    const float depth = pcz;
    const float zs = (depth > 1e-6f) ? depth : 1e-6f;
    const float u0 = fx * pcx / zs + cx;
    const float u1 = fy * pcy / zs + cy;

    // quaternion -> rotation, scaled columns
    float qw = rots_raw[n * 4 + 0], qx = rots_raw[n * 4 + 1];
    float qy = rots_raw[n * 4 + 2], qz = rots_raw[n * 4 + 3];
    const float qn = 1.0f / sqrtf(qw * qw + qx * qx + qy * qy + qz * qz);
    qw *= qn; qx *= qn; qy *= qn; qz *= qn;
    const float sx = expf(scales_raw[n * 3 + 0]);
    const float sy = expf(scales_raw[n * 3 + 1]);
    const float sz = expf(scales_raw[n * 3 + 2]);

    const float r00 = 1.0f - 2.0f * (qy * qy + qz * qz);
    const float r01 = 2.0f * (qx * qy - qw * qz);
    const float r02 = 2.0f * (qx * qz + qw * qy);
    const float r10 = 2.0f * (qx * qy + qw * qz);
    const float r11 = 1.0f - 2.0f * (qx * qx + qz * qz);
    const float r12 = 2.0f * (qy * qz - qw * qx);
    const float r20 = 2.0f * (qx * qz - qw * qy);
    const float r21 = 2.0f * (qy * qz + qw * qx);
    const float r22 = 1.0f - 2.0f * (qx * qx + qy * qy);

    const float m00 = r00 * sx, m01 = r01 * sy, m02 = r02 * sz;
    const float m10 = r10 * sx, m11 = r11 * sy, m12 = r12 * sz;
    const float m20 = r20 * sx, m21 = r21 * sy, m22 = r22 * sz;

    // cov3d = M * M^T (symmetric)
    const float c300 = m00 * m00 + m01 * m01 + m02 * m02;
    const float c301 = m00 * m10 + m01 * m11 + m02 * m12;
    const float c302 = m00 * m20 + m01 * m21 + m02 * m22;
    const float c311 = m10 * m10 + m11 * m11 + m12 * m12;
    const float c312 = m10 * m20 + m11 * m21 + m12 * m22;
    const float c322 = m20 * m20 + m21 * m21 + m22 * m22;

    // Jacobian with clipped tangent-plane coords
    const float limx = 1.3f * ((float)WIMG / (2.0f * fx));
    const float limy = 1.3f * ((float)HIMG / (2.0f * fy));
    const float txn = fminf(fmaxf(pcx / zs, -limx), limx) * zs;
    const float tyn = fminf(fmaxf(pcy / zs, -limy), limy) * zs;
    const float jA = fx / zs;
    const float jC = -fx * txn / (zs * zs);
    const float jB = fy / zs;
    const float jD = -fy * tyn / (zs * zs);

    // T = J @ Rcw   (2x3)
    const float T00 = jA * R00 + jC * R20;
    const float T01 = jA * R01 + jC * R21;
    const float T02 = jA * R02 + jC * R22;
    const float T10 = jB * R10 + jD * R20;
    const float T11 = jB * R11 + jD * R21;
    const float T12 = jB * R12 + jD * R22;

    // TC = T @ cov3d, cov2d = TC @ T^T
    const float TC00 = T00 * c300 + T01 * c301 + T02 * c302;
    const float TC01 = T00 * c301 + T01 * c311 + T02 * c312;
    const float TC02 = T00 * c302 + T01 * c312 + T02 * c322;
    const float TC10 = T10 * c300 + T11 * c301 + T12 * c302;
    const float TC11 = T10 * c301 + T11 * c311 + T12 * c312;
    const float TC12 = T10 * c302 + T11 * c312 + T12 * c322;

    const float a  = TC00 * T00 + TC01 * T01 + TC02 * T02 + 0.3f;
    const float b  = TC00 * T10 + TC01 * T11 + TC02 * T12;
    const float c2 = TC10 * T10 + TC11 * T11 + TC12 * T12 + 0.3f;
    const float det = a * c2 - b * b;
    const float cinv0 = c2 / det, cinv1 = -b / det, cinv2 = a / det;

    const float mid = 0.5f * (a + c2);
    const float lam = mid + sqrtf(fmaxf(mid * mid - det, 0.1f));
    const float radius = ceilf(3.0f * sqrtf(lam));

    // view direction from camera center
    const float twcx = -(R00 * t0 + R10 * t1 + R20 * t2);
    const float twcy = -(R01 * t0 + R11 * t1 + R21 * t2);
    const float twcz = -(R02 * t0 + R12 * t1 + R22 * t2);
    float dxv = pwx - twcx, dyv = pwy - twcy, dzv = pwz - twcz;
    const float dinv = 1.0f / sqrtf(dxv * dxv + dyv * dyv + dzv * dzv);
    const float x = dxv * dinv, y = dyv * dinv, z = dzv * dinv;
    const float xx = x * x, yy = y * y, zz = z * z;
    const float xy = x * y, yz = y * z, xzv = x * z;

    float bas[16];
    bas[0]  = 0.28209479177387814f;
    bas[1]  = -0.4886025119029199f * y;
    bas[2]  = 0.4886025119029199f * z;
    bas[3]  = -0.4886025119029199f * x;
    bas[4]  = 1.0925484305920792f * xy;
    bas[5]  = -1.0925484305920792f * yz;
    bas[6]  = 0.31539156525252005f * (2.0f * zz - xx - yy);
    bas[7]  = -1.0925484305920792f * xzv;
    bas[8]  = 0.5462742152960396f * (xx - yy);
    bas[9]  = -0.5900435899266435f * y * (3.0f * xx - yy);
    bas[10] = 2.890611442640554f * xy * z;
    bas[11] = -0.4570457994644658f * y * (4.0f * zz - xx - yy);
    bas[12] = 0.3731763325901154f * z * (2.0f * zz - 3.0f * xx - 3.0f * yy);
    bas[13] = -0.4570457994644658f * x * (4.0f * zz - xx - yy);
    bas[14] = 1.445305721320277f * z * (xx - yy);
    bas[15] = -0.5900435899266435f * x * (xx - 3.0f * yy);

    float col[3];
    #pragma unroll
    for (int ch = 0; ch < 3; ++ch) {
        float c = bas[0] * low_shs[n * 3 + ch];
        #pragma unroll
        for (int k = 1; k < 16; ++k)
            c += bas[k] * high_shs[n * 45 + (k - 1) * 3 + ch];
        col[ch] = fmaxf(c + 0.5f, 0.0f);
    }

    const float alpha = 1.0f / (1.0f + expf(-alphas_raw[n]));
    const float alphaEff = (depth > 0.2f) ? alpha : 0.0f;   // valid mask folded in

    float* r = rec + n * RECF;
    r[0] = u0; r[1] = u1;
    r[2] = cinv0; r[3] = cinv1; r[4] = cinv2;
    r[5] = alphaEff;
    r[6] = col[0]; r[7] = col[1]; r[8] = col[2];
    r[9] = 0.0f; r[10] = 0.0f; r[11] = 0.0f;

    depthOut[n] = depth;
    areasOut[n * 2 + 0] = radius;
    areasOut[n * 2 + 1] = radius;
}

// ---------------------------------------------------------------------------
// Kernel 2: single-block bitonic sort of 4096 (depth, index) pairs in LDS.
// Depth keys staged via ASYNCcnt-tracked global->LDS copies (no VGPR trip).
// ---------------------------------------------------------------------------
__global__ void __launch_bounds__(1024) sort_kernel(
    const float* __restrict__ depth, int* __restrict__ idxOut)
{
    __shared__ float sk[NPTS];
    __shared__ int   sv[NPTS];
    const int tid = threadIdx.x;

    // 1024 lanes x B128 = 16 KB of keys, DMA'd straight into LDS
    {
        const unsigned dst = lds_off_u32(&sk[0]) + (unsigned)tid * 16u;
        const unsigned off = (unsigned)tid * 16u;
        asm volatile("global_load_async_to_lds_b128 %0, %1, %2"
                     :: "v"(dst), "v"(off), "s"(depth) : "memory");
    }
    for (int i = tid; i < NPTS; i += 1024) sv[i] = i;
    wait_asynccnt0();
    __syncthreads();

    for (int k = 2; k <= NPTS; k <<= 1) {
        for (int j = k >> 1; j > 0; j >>= 1) {
            for (int i = tid; i < NPTS; i += 1024) {
                const int ixj = i ^ j;
                if (ixj > i) {
                    const bool up = ((i & k) == 0);
                    const float ki = sk[i], kj = sk[ixj];
                    if (up ? (ki > kj) : (ki < kj)) {
                        sk[i] = kj; sk[ixj] = ki;
                        const int t = sv[i]; sv[i] = sv[ixj]; sv[ixj] = t;
                    }
                }
            }
            __syncthreads();
        }
    }
    for (int i = tid; i < NPTS; i += 1024) idxOut[i] = sv[i];
}

// ---------------------------------------------------------------------------
// Kernel 3: gather records into sorted order (3 x float4 per record).
// ---------------------------------------------------------------------------
__global__ void __launch_bounds__(256) gather_kernel(
    const float4* __restrict__ rec, const int* __restrict__ idx,
    float4* __restrict__ recS)
{
    const int i = blockIdx.x * 256 + threadIdx.x;
    if (i >= NPTS) return;
    const int s = idx[i];
    recS[i * 3 + 0] = rec[s * 3 + 0];
    recS[i * 3 + 1] = rec[s * 3 + 1];
    recS[i * 3 + 2] = rec[s * 3 + 2];
}

// ---------------------------------------------------------------------------
// Kernel 4: tiled rasterizer.
//   - 1 lane = 1 pixel (wave32 -> 32 pixels/wave, 256 pixels/block, 36 blocks)
//   - Gaussian stream DMA'd into LDS by the TDM (tensor_load_to_lds),
//     double-buffered, TENSORcnt-synchronized; issuing wave = wave 0
//   - per-pixel sequential transmittance in a VGPR
//   - color accumulation offloaded to the matrix pipe:
//       D(16 ch x 16 px) += A(16 ch x 4 g) * B(4 g x 16 px), f32 WMMA 16x16x4
// ---------------------------------------------------------------------------
__global__ void __launch_bounds__(256) rasterize_kernel(
    const float* __restrict__ recS, float* __restrict__ image)
{
    __shared__ float gsm[2][16 * RECF];      // double-buffered gaussian chunk
    __shared__ float wsm[8][16 * WSTRIDE];   // per-wave weight tile (16 g x 32 px)

    const int tid  = threadIdx.x;
    const int wave = tid >> 5;
    const int lane = tid & 31;
    const int m    = lane & 15;      // WMMA row (channel) / tile column
    const int half = lane >> 4;      // selects K pair {0,1} vs {2,3}
    float* ws = wsm[wave];

    const int pixBase = blockIdx.x * 256 + wave * 32;
    const int p = pixBase + lane;
    const float px = (float)(p % WIMG);
    const float py = (float)(p / WIMG);

    float tau = 1.0f;
    v8f acc0 = {0.f,0.f,0.f,0.f,0.f,0.f,0.f,0.f};
    v8f acc1 = {0.f,0.f,0.f,0.f,0.f,0.f,0.f,0.f};

    // prologue: TDM-stage chunk 0
    if (wave == 0) tdm_load_chunk(lds_off_u32(&gsm[0][0]), recS);

    const int NCHUNK = NPTS / 16;    // 256
    for (int it = 0; it < NCHUNK; ++it) {
        if (wave == 0) wait_tensorcnt0();   // staged chunk landed in LDS
        __syncthreads();                    // publish; prev WMMA reads done
        const float* gr = gsm[it & 1];

        // kick DMA for the next chunk into the other buffer
        if (wave == 0 && it + 1 < NCHUNK)
            tdm_load_chunk(lds_off_u32(&gsm[(it + 1) & 1][0]),
                           recS + (it + 1) * (16 * RECF));
        if (it + 2 < NCHUNK)
            __builtin_prefetch(&recS[(it + 2) * (16 * RECF)], 0, 1);

        // sequential alpha compositing for this lane's pixel over 16 gaussians
        #pragma unroll 4
        for (int k = 0; k < 16; ++k) {
            const float* g = gr + k * RECF;
            const float dx = g[0] - px;
            const float dy = g[1] - py;
            float pw = -0.5f * (g[2] * dx * dx + g[4] * dy * dy) - g[3] * dx * dy;
            float ap = g[5] * expf(fminf(pw, 0.0f));
            ap = fminf(ap, 0.99f);
            ap = (ap >= (1.0f / 255.0f)) ? ap : 0.0f;
            const float w = (tau > 1e-4f) ? ap * tau : 0.0f;
            tau *= (1.0f - ap);
            ws[k * WSTRIDE + lane] = w;
        }
        __syncthreads();             // weight tile visible wave/block-wide

        // matrix-pipe color accumulation: 4 groups of 4 gaussians, 2 pixel tiles
        #pragma unroll
        for (int g = 0; g < 4; ++g) {
            const int kA = g * 4 + half * 2;
            const int ci = kA * RECF + 6 + ((m < 3) ? m : 0);
            float a0 = gr[ci];            a0 = (m < 3) ? a0 : 0.0f;
            float a1 = gr[ci + RECF];     a1 = (m < 3) ? a1 : 0.0f;
            v2f A = {a0, a1};
            v2f B0 = {ws[kA * WSTRIDE + m],      ws[(kA + 1) * WSTRIDE + m]};
            v2f B1 = {ws[kA * WSTRIDE + 16 + m], ws[(kA + 1) * WSTRIDE + 16 + m]};
            acc0 = __builtin_amdgcn_wmma_f32_16x16x4_f32(
                false, A, false, B0, 0, acc0, false, false);
            acc1 = __builtin_amdgcn_wmma_f32_16x16x4_f32(
                false, A, false, B1, 0, acc1, false, false);
        }
    }

    // D rows 0..2 (RGB) live in acc[0..2] of lanes 0..15
    if (lane < 16) {
        #pragma unroll
        for (int c = 0; c < 3; ++c) {
            image[c * NPIX + pixBase + lane]      = acc0[c];
            image[c * NPIX + pixBase + 16 + lane] = acc1[c];
        }
    }
}

// ---------------------------------------------------------------------------
extern "C" void kernel_launch(void* const* d_in, const int* in_sizes, int n_in,
                              void* d_out, int out_size, void* d_ws, size_t ws_size,
                              hipStream_t stream)
{
    (void)in_sizes; (void)n_in; (void)out_size; (void)ws_size;
    const float* pws        = (const float*)d_in[0];
    const float* low_shs    = (const float*)d_in[1];
    const float* high_shs   = (const float*)d_in[2];
    const float* alphas_raw = (const float*)d_in[3];
    const float* scales_raw = (const float*)d_in[4];
    const float* rots_raw   = (const float*)d_in[5];
    // d_in[6] = us (unused by reference math)
    const float* Rcw        = (const float*)d_in[7];
    const float* tcw        = (const float*)d_in[8];
    const float* cam        = (const float*)d_in[9];

    float* out   = (float*)d_out;
    float* image = out;                  // 3 * 9216 floats
    float* areas = out + 3 * NPIX;       // 4096 * 2 floats

    char* ws = (char*)d_ws;
    float* rec   = (float*)(ws + 0);        // 4096*12 floats = 196608 B
    float* recS  = (float*)(ws + 196608);   // 196608 B
    float* depth = (float*)(ws + 393216);   // 16384 B
    int*   idx   = (int*)  (ws + 409600);   // 16384 B

    preprocess_kernel<<<NPTS / 256, 256, 0, stream>>>(
        pws, low_shs, high_shs, alphas_raw, scales_raw, rots_raw,
        Rcw, tcw, cam, rec, depth, areas);
    sort_kernel<<<1, 1024, 0, stream>>>(depth, idx);
    gather_kernel<<<NPTS / 256, 256, 0, stream>>>(
        (const float4*)rec, idx, (float4*)recS);
    rasterize_kernel<<<NPIX / 256, 256, 0, stream>>>(recS, image);
}